// RNN_Hierarchical_7928509628601
// MI455X (gfx1250) — compile-verified
//
#include <hip/hip_runtime.h>
#include <stdint.h>

// ---------------------------------------------------------------------------
// Persistent fp32-WMMA RNN for gfx1250 (MI455X).
//   T=2048, B=256, I=64, N=256, C=2, tau=2 -> decay=0.5, 1/tau=0.5
// Grid: 16 workgroups, each owns a 16-row batch tile for all 2048 steps.
// W_hh (scaled, diag-zeroed, k-pair interleaved) lives in LDS the whole time.
// GEMMs use V_WMMA_F32_16X16X4_F32 (exact fp32, matching the reference).
// x_t tiles are double-buffered via async global->LDS DMA (ASYNCcnt).
// ---------------------------------------------------------------------------

typedef float v2f __attribute__((ext_vector_type(2)));
typedef float v8f __attribute__((ext_vector_type(8)));

#define T_STEPS   2048
#define BATCH     256
#define ISZ       64
#define NSZ       256
#define CSZ       2
#define BT        16          // batch rows per workgroup
#define NBLK      (BATCH/BT)  // 16 workgroups
#define INV_TAU   0.5f
#define DECAY     0.5f
#define NSLOPE    0.01f

// LDS layout (floats):
//  WB  : W_hh^T scaled, stored as [k/2][n][k&1], n-stride 272 -> row = 544 f
//        (544 % 64 == 32 and n*2 step -> conflict-free b64 B-frag loads)
#define WB_ROW    544
#define WB_FLOATS (128 * WB_ROW)        // 69632 f = 272 KB
//  hbuf: h state [BT][260]  (stride 260 -> 4*lane bank step, conflict-free)
#define H_STRIDE  260
#define H_FLOATS  (BT * H_STRIDE)       // 4160 f
//  xbuf[2]: double-buffered raw data_t tiles, [BT][68] each
#define XT_STRIDE 68
#define XT_FLOATS (BT * XT_STRIDE)      // 1088 f
#define SMEM_FLOATS (WB_FLOATS + H_FLOATS + 2 * XT_FLOATS)  // 75968
#define SMEM_BYTES  (SMEM_FLOATS * 4)                       // 303872 B < 320 KB

__device__ __forceinline__ v8f wmma4(v2f a, v2f b, v8f c) {
  // D = A(16x4,f32) * B(4x16,f32) + C(16x16,f32)
  return __builtin_amdgcn_wmma_f32_16x16x4_f32(
      /*neg_a=*/false, a, /*neg_b=*/false, b,
      /*c_mod=*/(short)0, c, /*reuse_a=*/false, /*reuse_b=*/false);
}

// Async DMA of one 16x64 f32 tile (contiguous 4096 B in global) into LDS with
// row padding to XT_STRIDE. One b128 per lane of a 256-thread block.
__device__ __forceinline__ void async_stage_x(uint32_t lds_byte_off,
                                              const float* gbase, int tid) {
  // lane's source: bytes [tid*16, tid*16+16); dest row m=tid>>4, col (tid&15)*4
  uint32_t dst = lds_byte_off + (uint32_t)((tid >> 4) * (XT_STRIDE * 4) +
                                           (tid & 15) * 16);
  uint32_t voff = (uint32_t)(tid * 16);
  uint64_t saddr = (uint64_t)(uintptr_t)gbase;
  asm volatile("global_load_async_to_lds_b128 %0, %1, %2 offset:0"
               :: "v"(dst), "v"(voff), "s"(saddr)
               : "memory");
}

__global__ __launch_bounds__(256, 1)
void rnn_hier_persistent(const float* __restrict__ data,
                         const float* __restrict__ hier,
                         const float* __restrict__ h0,
                         const float* __restrict__ W_in,
                         const float* __restrict__ b_in,
                         const float* __restrict__ W_hh,
                         const float* __restrict__ b_hh,
                         const float* __restrict__ W_ff,
                         const float* __restrict__ b_ff,
                         const float* __restrict__ W_fc,
                         const float* __restrict__ b_fc,
                         float* __restrict__ out) {
  extern __shared__ float smem[];
  float* WB   = smem;
  float* hbuf = smem + WB_FLOATS;
  float* xbuf0 = hbuf + H_FLOATS;
  float* xbuf1 = xbuf0 + XT_FLOATS;

  const int tid  = threadIdx.x;
  const int lane = tid & 31;
  const int wv   = tid >> 5;        // 8 waves
  const int half = lane >> 4;       // ISA 16x4 A layout: lanes 16-31 -> K+2
  const int nin  = lane & 15;
  const int ma   = lane & 15;       // A-operand row (M)
  const int wgB0 = blockIdx.x * BT;
  const int n0   = (wv * 2) * 16;   // this wave's two 16-wide N tiles
  const int n1   = n0 + 16;

  // LDS byte offsets of the two x buffers (flat ptr low 32 bits == LDS offset)
  const uint32_t xoff0 = (uint32_t)(uintptr_t)xbuf0;
  const uint32_t xoff1 = (uint32_t)(uintptr_t)xbuf1;

  // ---- Phase 1: stage hier rows, compute cb = (b_in+b_hh+ff)/tau in C-layout
  for (int idx = tid; idx < BT * NSZ; idx += 256) {
    int m = idx >> 8, k = idx & 255;
    hbuf[m * H_STRIDE + k] = hier[(size_t)(wgB0 + m) * NSZ + k];
  }
  __syncthreads();

  float cb0[8], cb1[8];
  {
    float a0[8], a1[8];
#pragma unroll
    for (int r = 0; r < 8; ++r) { a0[r] = 0.f; a1[r] = 0.f; }
    const float* wf0 = W_ff + (size_t)(n0 + nin) * NSZ;
    const float* wf1 = W_ff + (size_t)(n1 + nin) * NSZ;
    for (int j = 0; j < NSZ; ++j) {
      float w0 = wf0[j], w1 = wf1[j];
#pragma unroll
      for (int r = 0; r < 8; ++r) {
        float hv = hbuf[(r + 8 * half) * H_STRIDE + j];  // LDS broadcast
        a0[r] += hv * w0;
        a1[r] += hv * w1;
      }
    }
    const int nA = n0 + nin, nB = n1 + nin;
    const float cA = b_ff[nA] + b_in[nA] + b_hh[nA];
    const float cB = b_ff[nB] + b_in[nB] + b_hh[nB];
#pragma unroll
    for (int r = 0; r < 8; ++r) {
      cb0[r] = (a0[r] + cA) * INV_TAU;
      cb1[r] = (a1[r] + cB) * INV_TAU;
    }
  }
  __syncthreads();  // done reading staged hier from hbuf

  // ---- Phase 2: build LDS-resident W_hh^T (scaled, diag zero), init h
  for (int idx = tid; idx < NSZ * NSZ; idx += 256) {
    int n = idx >> 8, k = idx & 255;                 // coalesced W_hh read
    float v = W_hh[idx] * ((n == k) ? 0.f : INV_TAU);
    WB[(k >> 1) * WB_ROW + (n << 1) + (k & 1)] = v;  // B[k][n], pair-interleaved
  }
  for (int idx = tid; idx < BT * NSZ; idx += 256) {
    int m = idx >> 8, k = idx & 255;
    hbuf[m * H_STRIDE + k] = h0[(size_t)(wgB0 + m) * NSZ + k];
  }
  float h0f[8], h1f[8];
#pragma unroll
  for (int r = 0; r < 8; ++r) {
    int m = r + 8 * half;  // accumulator layout: VGPR r -> M=r / M=r+8
    h0f[r] = h0[(size_t)(wgB0 + m) * NSZ + n0 + nin];
    h1f[r] = h0[(size_t)(wgB0 + m) * NSZ + n1 + nin];
  }

  // prefetch x_0 into buffer 0 (ASYNCcnt = 1)
  async_stage_x(xoff0, data + (size_t)wgB0 * ISZ, tid);
  __syncthreads();

  // ---- Phase 3: the 2048-step recurrence, everything hot in LDS/VGPRs
  const float* hA   = hbuf + ma * H_STRIDE + 2 * half;
  const float* wbB0 = WB + half * WB_ROW + ((n0 + nin) << 1);
  const float* wbB1 = WB + half * WB_ROW + ((n1 + nin) << 1);
  const float* wB0  = W_in + (size_t)(n0 + nin) * ISZ + 2 * half;
  const float* wB1  = W_in + (size_t)(n1 + nin) * ISZ + 2 * half;
  const uint32_t xAoff = (uint32_t)(ma * XT_STRIDE + 2 * half);

  for (int t = 0; t < T_STEPS; ++t) {
    // kick off async fetch of x_{t+1} into the other buffer (overlaps GEMM)
    {
      int tn = (t + 1 < T_STEPS) ? (t + 1) : t;  // clamped dummy on last step
      uint32_t doff = ((t + 1) & 1) ? xoff1 : xoff0;
      async_stage_x(doff, data + ((size_t)tn * BATCH + wgB0) * ISZ, tid);
    }
    // wait for x_t (issued last step); newest fetch may stay outstanding
    asm volatile("s_wait_asynccnt 0x1" ::: "memory");
    __syncthreads();  // all waves' x_t slices + previous h writes visible

    v8f acc0 = {0.f, 0.f, 0.f, 0.f, 0.f, 0.f, 0.f, 0.f};
    v8f acc1 = {0.f, 0.f, 0.f, 0.f, 0.f, 0.f, 0.f, 0.f};
    v8f axc0 = {0.f, 0.f, 0.f, 0.f, 0.f, 0.f, 0.f, 0.f};
    v8f axc1 = {0.f, 0.f, 0.f, 0.f, 0.f, 0.f, 0.f, 0.f};

    // h @ (W_hh^T / tau): K = 256, fp32 WMMA, operands from LDS
    {
      const float* ap  = hA;
      const float* bp0 = wbB0;
      const float* bp1 = wbB1;
#pragma unroll 4
      for (int k0 = 0; k0 < NSZ; k0 += 4) {
        v2f a  = *(const v2f*)(ap + k0);
        v2f b0 = *(const v2f*)bp0;
        v2f b1 = *(const v2f*)bp1;
        bp0 += 2 * WB_ROW;
        bp1 += 2 * WB_ROW;
        acc0 = wmma4(a, b0, acc0);
        acc1 = wmma4(a, b1, acc1);
      }
    }
    // x_t @ W_in^T (unscaled): K = 64, B frags stream from L1-resident W_in
    {
      const float* xA = (const float*)((t & 1) ? xbuf1 : xbuf0) + xAoff;
#pragma unroll 4
      for (int k0 = 0; k0 < ISZ; k0 += 4) {
        v2f a  = *(const v2f*)(xA + k0);
        v2f b0 = *(const v2f*)(wB0 + k0);
        v2f b1 = *(const v2f*)(wB1 + k0);
        axc0 = wmma4(a, b0, axc0);
        axc1 = wmma4(a, b1, axc1);
      }
    }

    // elementwise: h = leaky_relu(decay*h + acc_hh + acc_x/tau + cb)
#pragma unroll
    for (int r = 0; r < 8; ++r) {
      float p0 = DECAY * h0f[r] + acc0[r] + INV_TAU * axc0[r] + cb0[r];
      float p1 = DECAY * h1f[r] + acc1[r] + INV_TAU * axc1[r] + cb1[r];
      h0f[r] = (p0 >= 0.f) ? p0 : NSLOPE * p0;
      h1f[r] = (p1 >= 0.f) ? p1 : NSLOPE * p1;
    }
    __syncthreads();  // all A-operand reads of hbuf (and xbuf) complete
#pragma unroll
    for (int r = 0; r < 8; ++r) {
      int m = r + 8 * half;
      hbuf[m * H_STRIDE + n0 + nin] = h0f[r];
      hbuf[m * H_STRIDE + n1 + nin] = h1f[r];
    }
  }
  __syncthreads();

  // ---- Phase 4: readout out[b, 0:2] = h_final @ W_fc^T + b_fc (tiny)
  if (tid < BT * CSZ) {
    int m = tid >> 1, c = tid & 1;
    float s = b_fc[c];
    const float* wr = W_fc + (size_t)c * NSZ;
    const float* hr = hbuf + m * H_STRIDE;
    for (int j = 0; j < NSZ; ++j) s += hr[j] * wr[j];
    out[(size_t)(wgB0 + m) * CSZ + c] = s;
  }
}

extern "C" void kernel_launch(void* const* d_in, const int* in_sizes, int n_in,
                              void* d_out, int out_size, void* d_ws, size_t ws_size,
                              hipStream_t stream) {
  (void)in_sizes; (void)n_in; (void)out_size; (void)d_ws; (void)ws_size;
  const float* data = (const float*)d_in[0];
  const float* hier = (const float*)d_in[1];
  const float* h0   = (const float*)d_in[2];
  const float* W_in = (const float*)d_in[3];
  const float* b_in = (const float*)d_in[4];
  const float* W_hh = (const float*)d_in[5];
  const float* b_hh = (const float*)d_in[6];
  const float* W_ff = (const float*)d_in[7];
  const float* b_ff = (const float*)d_in[8];
  const float* W_fc = (const float*)d_in[9];
  const float* b_fc = (const float*)d_in[10];
  float* out = (float*)d_out;

  // allow ~300 KB dynamic LDS (CDNA5 WGP has 320 KB)
  (void)hipFuncSetAttribute((const void*)rnn_hier_persistent,
                            hipFuncAttributeMaxDynamicSharedMemorySize,
                            SMEM_BYTES);

  rnn_hier_persistent<<<NBLK, 256, SMEM_BYTES, stream>>>(
      data, hier, h0, W_in, b_in, W_hh, b_hh, W_ff, b_ff, W_fc, b_fc, out);
}